// Fourier_36189394436368
// MI455X (gfx1250) — compile-verified
//
#include <hip/hip_runtime.h>

typedef float v2f __attribute__((ext_vector_type(2)));
typedef float v8f __attribute__((ext_vector_type(8)));

// ---------------------------------------------------------------------------
// Kernel 1: PAB[512,128] = X[512,1024] @ W[1024,128], where W = [A | B]
// (A = a[:, :, 0], B = b[:, :, 0], both [1024, 64]).
// One wave32 per 16x16 output tile; K-reduction via V_WMMA_F32_16X16X4_F32.
//
// f32 16x4 A-operand layout (ISA 7.12.2): lane l holds row M = l%16,
// VGPR0/1 = K = k0+koff, k0+koff+1 with koff = 2*(l>=16).  B (4x16) mirrors
// this: lane l holds col N = l%16, rows k0+koff and k0+koff+1.
// C/D: VGPR v holds M = v + 8*(l>=16), N = l%16.
// ---------------------------------------------------------------------------
__global__ __launch_bounds__(32)
void fourier_coeff_gemm(const float* __restrict__ X,
                        const float* __restrict__ A,
                        const float* __restrict__ Bm,
                        float* __restrict__ PAB)
{
    const int ntile = blockIdx.x;              // 0..7   (128 cols / 16)
    const int mtile = blockIdx.y;              // 0..31  (512 rows / 16)
    const int lane  = threadIdx.x;             // 0..31
    const int m0    = mtile * 16;
    const int n0    = ntile * 16;
    const int row   = lane & 15;
    const int koff  = (lane >> 4) * 2;

    // Each 16-wide n-tile lies entirely inside A (cols 0..63) or B (64..127).
    const float* __restrict__ Wsrc = (n0 < 64) ? A : Bm;
    const int wcol = (n0 & 63) + row;          // column within A or B

    v8f acc = {};
    for (int k0 = 0; k0 < 1024; k0 += 4) {
        // A operand: two consecutive K elements of row (m0+row) -> b64 load
        v2f av = *(const v2f*)(X + (size_t)(m0 + row) * 1024 + (k0 + koff));
        // B operand: rows k0+koff, k0+koff+1 of W, column wcol
        v2f bv;
        bv.x = Wsrc[(size_t)(k0 + koff)     * 64 + wcol];
        bv.y = Wsrc[(size_t)(k0 + koff + 1) * 64 + wcol];
        acc = __builtin_amdgcn_wmma_f32_16x16x4_f32(
            /*neg_a=*/false, av, /*neg_b=*/false, bv,
            /*c_mod=*/(short)0, acc, /*reuse_a=*/false, /*reuse_b=*/false);
    }

    const int mo = (lane >> 4) * 8;
    #pragma unroll
    for (int v = 0; v < 8; ++v) {
        PAB[(size_t)(m0 + mo + v) * 128 + n0 + row] = acc[v];
    }
}

// ---------------------------------------------------------------------------
// Kernel 2: out[b,p] = a0/(2P) + sum_k pa[b,k] cos(k*theta) + pb[b,k] sin(k*theta),
// theta = (2*pi/P) * x[b,p].  One sincos per element + angle-addition
// recurrence: cos/sin(k*theta) rotated by (cs, sn) each step (4 FMA/k),
// plus 2 accumulation FMAs/k.  pa/pb row addresses are wave-uniform
// (b = blockIdx.y) -> scalar loads.
// ---------------------------------------------------------------------------
__global__ __launch_bounds__(256)
void fourier_eval(const float* __restrict__ X,
                  const float* __restrict__ PAB,
                  const float* __restrict__ a0,
                  float* __restrict__ out)
{
    const int p = blockIdx.x * 256 + threadIdx.x;   // 0..1023
    const int b = blockIdx.y;                        // 0..511

    const float two_pi_over_P = 6.28318530717958647692f / 1024.0f;
    const float xv    = X[(size_t)b * 1024 + p];
    const float theta = two_pi_over_P * xv;

    float sn, cs;
    __sincosf(theta, &sn, &cs);                      // v_sin_f32 / v_cos_f32

    const float* __restrict__ pa = PAB + (size_t)b * 128;  // uniform row
    const float* __restrict__ pb = pa + 64;

    // k = 0 term: cos=1, sin=0
    float acc = a0[0] * (1.0f / 2048.0f) + pa[0];

    float ck = 1.0f, sk = 0.0f;                      // cos/sin(k*theta)
    #pragma unroll
    for (int k = 1; k < 64; ++k) {
        float cn = ck * cs - sk * sn;
        float s2 = sk * cs + ck * sn;
        acc = fmaf(pa[k], cn, acc);
        acc = fmaf(pb[k], s2, acc);
        ck = cn;
        sk = s2;
    }

    out[(size_t)b * 1024 + p] = acc;
}

extern "C" void kernel_launch(void* const* d_in, const int* in_sizes, int n_in,
                              void* d_out, int out_size, void* d_ws, size_t ws_size,
                              hipStream_t stream) {
    const float* x  = (const float*)d_in[0];   // [512, 1024]
    const float* a  = (const float*)d_in[1];   // [1024, 64, 1]
    const float* b  = (const float*)d_in[2];   // [1024, 64, 1]
    const float* a0 = (const float*)d_in[3];   // [1]
    float* pab = (float*)d_ws;                 // [512, 128] = 256 KB scratch
    float* out = (float*)d_out;                // [512, 1024]

    dim3 g1(8, 32);                            // n-tiles x m-tiles
    fourier_coeff_gemm<<<g1, dim3(32), 0, stream>>>(x, a, b, pab);

    dim3 g2(4, 512);                           // p-tiles x batch
    fourier_eval<<<g2, dim3(256), 0, stream>>>(x, pab, a0, out);
}